// ParacrineCascade_47253230190597
// MI455X (gfx1250) — compile-verified
//
#include <hip/hip_runtime.h>

typedef __attribute__((ext_vector_type(2))) float v2f;
typedef __attribute__((ext_vector_type(4))) float v4f;
typedef __attribute__((ext_vector_type(8))) float v8f;
typedef __attribute__((ext_vector_type(4))) unsigned int u32x4;
typedef __attribute__((ext_vector_type(8))) int i32x8;
typedef __attribute__((ext_vector_type(4))) int i32x4;

// Problem shape from the reference (fixed).
#define BB 32
#define NN 2048
#define DD 128
#define WPB 4            // waves per block
#define NTILES (NN/16)   // 128 column tiles
#define TILE_DW   (16 * DD)    // 2048 dwords = 8KB per B tile (contiguous in global)
#define ROW_PITCH (DD + 4)     // 132 dwords: TDM pad gives conflict-free LDS reads

// ---------------------------------------------------------------------------
// Kernel 1: per-row squared norms |x_n|^2  ->  ws (B*N floats)
// ---------------------------------------------------------------------------
__global__ void pc_norms_kernel(const float* __restrict__ x,
                                float* __restrict__ norms) {
  int row = blockIdx.x * blockDim.x + threadIdx.x;
  if (row >= BB * NN) return;
  const float* p = x + (size_t)row * DD;
  float s = 0.f;
#pragma unroll
  for (int i = 0; i < DD / 4; ++i) {
    v4f v = *(const v4f*)(p + 4 * i);
    s += v[0] * v[0] + v[1] * v[1] + v[2] * v[2] + v[3] * v[3];
  }
  norms[row] = s;
}

// ---------------------------------------------------------------------------
// TDM: issue one descriptor copying a contiguous 8KB B tile (2048 f32) from
// global to LDS, inserting 4 dwords of padding every 128 dwords so each
// 16-row tile lands at a 132-dword row pitch (bank-conflict-free fragments).
// D# layout per CDNA5 ISA ch.8 (groups 0/1; groups 2/3 zero/unused, <=2D).
// ---------------------------------------------------------------------------
__device__ __forceinline__ void tdm_load_tile(const float* gsrc, unsigned lds_off) {
  unsigned long long ga = (unsigned long long)(uintptr_t)gsrc;
  u32x4 g0;
  g0[0] = 1u;                                   // count=1, user descriptor
  g0[1] = lds_off;                              // LDS destination (bytes)
  g0[2] = (unsigned)ga;                         // global_addr[31:0]
  g0[3] = (unsigned)((ga >> 32) & 0x1FFFFFFu)   // global_addr[56:32]
        | (2u << 30);                           // type = 2 ("image")
  i32x8 g1;
  g1[0] = (2 << 16)                             // data_size = 4 bytes
        | (1 << 20)                             // pad_enable
        | (6 << 22)                             // pad_interval: every 128 dwords
        | (3 << 25);                            // pad_amount: 4 dwords
  g1[1] = (int)((unsigned)TILE_DW << 16);       // tensor_dim0 lo16 (one 2048-elem line)
  g1[2] = (int)(1u << 16);                      // tensor_dim0 hi16=0 | tensor_dim1=1
  g1[3] = (int)((unsigned)TILE_DW << 16);       // tensor_dim1 hi16=0 | tile_dim0=2048
  g1[4] = 1;                                    // tile_dim1=1, tile_dim2=0(unused)
  g1[5] = TILE_DW;                              // tensor_dim0_stride lo32
  g1[6] = 0;                                    // stride hi | tensor_dim1_stride lo
  g1[7] = 0;
  i32x4 gz4 = {0, 0, 0, 0};                     // groups 2/3: unused (<=2D tensor)
  i32x8 gz8 = {0, 0, 0, 0, 0, 0, 0, 0};
  __builtin_amdgcn_tensor_load_to_lds(g0, g1, gz4, gz4, gz8, 0);
}

// ---------------------------------------------------------------------------
// Kernel 2: fused Gram (fp32 WMMA) + top-3 + gather + blend.
// One wave (32 lanes) owns a 16-row tile; the block's 4 waves share TDM-staged
// double-buffered B tiles in LDS.
// ---------------------------------------------------------------------------
__global__ __launch_bounds__(WPB * 32)
void pc_knnmix_kernel(const float* __restrict__ x,
                      const float* __restrict__ norms,
                      const float* __restrict__ strength_p,
                      float* __restrict__ out) {
  __shared__ float tbuf[2][16 * ROW_PITCH];     // 2 x 8448 B staged B tiles
  __shared__ float sd[WPB][16][48];
  __shared__ int   si[WPB][16][48];
  __shared__ int   stop3[WPB][16][3];

  const int wave = threadIdx.x >> 5;
  const int lane = threadIdx.x & 31;
  const int half = lane >> 4;   // 0: lanes 0-15, 1: lanes 16-31
  const int l16  = lane & 15;

  const int blocksPerBatch = NTILES / WPB;                 // 32
  const int b       = blockIdx.x / blocksPerBatch;
  const int rowTile = (blockIdx.x % blocksPerBatch) * WPB + wave;
  const int rowBase = rowTile * 16;

  const float* xb = x     + (size_t)b * NN * DD;
  const float* nb = norms + (size_t)b * NN;

  const unsigned lds0 = (unsigned)(uintptr_t)(&tbuf[0][0]);
  const unsigned ldsoff[2] = {lds0, lds0 + (unsigned)(16 * ROW_PITCH * 4)};

  // ---- A panel: 16 rows x 128 K, as 32 fragments of V_WMMA_F32_16X16X4_F32.
  // A 16x4 layout: all lanes carry row = l16; lanes 0-15 hold K={0,1},
  // lanes 16-31 hold K={2,3} of each 4-wide chunk (2 consecutive K / lane).
  v2f afrag[32];
  {
    const float* arow = xb + (size_t)(rowBase + l16) * DD + 2 * half;
#pragma unroll
    for (int c = 0; c < 32; ++c)
      afrag[c] = *(const v2f*)(arow + 4 * c);
  }

  // ---- per-lane top-3 state for 8 rows (C/D layout: lane holds column l16,
  // rows m = 8*half + v in accumulator VGPR v).
  float t0[8], t1[8], t2[8];
  int   j0[8], j1[8], j2[8];
#pragma unroll
  for (int v = 0; v < 8; ++v) {
    t0[v] = t1[v] = t2[v] = 3.0e38f;
    j0[v] = j1[v] = j2[v] = -1;
  }

  // Prologue: stage tile 0.
  if (wave == 0) tdm_load_tile(xb, ldsoff[0]);

  for (int jt = 0; jt < NTILES; ++jt) {
    if (wave == 0) {
      if (jt + 1 < NTILES) {
        tdm_load_tile(xb + (size_t)(jt + 1) * TILE_DW, ldsoff[(jt + 1) & 1]);
        __builtin_amdgcn_s_wait_tensorcnt(1);   // current tile done; next in flight
      } else {
        __builtin_amdgcn_s_wait_tensorcnt(0);
      }
    }
    __syncthreads();                            // tile jt visible to all waves

    // B 4x16 fragment from LDS: lane carries column = l16, same K split as A.
    // Row pitch 132 dwords -> 32 distinct banks across the wave.
    const float* tp = &tbuf[jt & 1][l16 * ROW_PITCH + 2 * half];

    v8f acc = {};
#pragma unroll
    for (int c = 0; c < 32; ++c) {
      v2f bf = *(const v2f*)(tp + 4 * c);
      acc = __builtin_amdgcn_wmma_f32_16x16x4_f32(
          /*neg_a=*/false, afrag[c], /*neg_b=*/false, bf,
          /*c_mod=*/(short)0, acc, /*reuse_a=*/false, /*reuse_b=*/false);
    }

    const int   colBase = jt * 16;
    const int   gn      = colBase + l16;
    const float keyb    = nb[gn];     // |x_n|^2 ; |x_m|^2 drops out of ranking
#pragma unroll
    for (int v = 0; v < 8; ++v) {
      const int gm = rowBase + 8 * half + v;
      if (gn == gm) continue;         // diagonal mask (+1e6 in reference)
      const float key = keyb - 2.0f * acc[v];
      if (key < t0[v]) {
        t2[v] = t1[v]; j2[v] = j1[v];
        t1[v] = t0[v]; j1[v] = j0[v];
        t0[v] = key;   j0[v] = gn;
      } else if (key < t1[v]) {
        t2[v] = t1[v]; j2[v] = j1[v];
        t1[v] = key;   j1[v] = gn;
      } else if (key < t2[v]) {
        t2[v] = key;   j2[v] = gn;
      }
    }
    __syncthreads();                  // all waves done with buf[jt&1] before reuse
  }

  // ---- dump per-lane top-3 to LDS (16 lanes contribute per row)
#pragma unroll
  for (int v = 0; v < 8; ++v) {
    const int r = 8 * half + v;
    sd[wave][r][l16 * 3 + 0] = t0[v]; si[wave][r][l16 * 3 + 0] = j0[v];
    sd[wave][r][l16 * 3 + 1] = t1[v]; si[wave][r][l16 * 3 + 1] = j1[v];
    sd[wave][r][l16 * 3 + 2] = t2[v]; si[wave][r][l16 * 3 + 2] = j2[v];
  }
  __syncthreads();

  // ---- per-row merge of 48 candidates (lanes 0..15 each own one row)
  if (lane < 16) {
    const int r = lane;
    float m0 = 3.0e38f, m1 = 3.0e38f, m2 = 3.0e38f;
    int   q0 = -1, q1 = -1, q2 = -1;
    for (int e = 0; e < 48; ++e) {
      const float d  = sd[wave][r][e];
      const int   ix = si[wave][r][e];
      if (ix < 0) continue;
      const bool lt0 = (d < m0) || (d == m0 && ix < q0);
      const bool lt1 = (d < m1) || (d == m1 && ix < q1);
      const bool lt2 = (d < m2) || (d == m2 && ix < q2);
      if (lt0)      { m2 = m1; q2 = q1; m1 = m0; q1 = q0; m0 = d; q0 = ix; }
      else if (lt1) { m2 = m1; q2 = q1; m1 = d;  q1 = ix; }
      else if (lt2) { m2 = d;  q2 = ix; }
    }
    stop3[wave][r][0] = q0;
    stop3[wave][r][1] = q1;
    stop3[wave][r][2] = q2;
  }
  __syncthreads();

  // ---- gather 3 neighbors, blend, write out
  float s = strength_p[0];
  s = fminf(fmaxf(s, 0.0f), 1.0f);
  const float w_self = 1.0f - s;
  const float w_nb   = s * (1.0f / 3.0f);

  float* ob = out + (size_t)b * NN * DD;
  const int d = lane * 4;             // 32 lanes x float4 = 128
  for (int r = 0; r < 16; ++r) {
    const int gm = rowBase + r;
    const int i0 = stop3[wave][r][0];
    const int i1 = stop3[wave][r][1];
    const int i2 = stop3[wave][r][2];
    v4f xs = *(const v4f*)(xb + (size_t)gm * DD + d);
    v4f n0 = *(const v4f*)(xb + (size_t)i0 * DD + d);
    v4f n1 = *(const v4f*)(xb + (size_t)i1 * DD + d);
    v4f n2 = *(const v4f*)(xb + (size_t)i2 * DD + d);
    v4f o;
#pragma unroll
    for (int q = 0; q < 4; ++q)
      o[q] = w_self * xs[q] + w_nb * (n0[q] + n1[q] + n2[q]);
    *(v4f*)(ob + (size_t)gm * DD + d) = o;
  }
}

// ---------------------------------------------------------------------------
extern "C" void kernel_launch(void* const* d_in, const int* in_sizes, int n_in,
                              void* d_out, int out_size, void* d_ws, size_t ws_size,
                              hipStream_t stream) {
  const float* features = (const float*)d_in[0];
  const float* strength = (const float*)d_in[1];
  float*       out      = (float*)d_out;
  float*       norms    = (float*)d_ws;   // B*N floats = 256 KB

  pc_norms_kernel<<<(BB * NN + 255) / 256, 256, 0, stream>>>(features, norms);

  pc_knnmix_kernel<<<BB * (NTILES / WPB), WPB * 32, 0, stream>>>(
      features, norms, strength, out);
}